// FPQuantizer_76312978915927
// MI455X (gfx1250) — compile-verified
//
#include <hip/hip_runtime.h>
#include <stdint.h>

// Per-row flex-FP8(e4m3) quantizer, 8192x8192 fp32.
// Memory-bound: ~537 MB traffic -> ~23us floor at 23.3 TB/s.
// One block per row; row staged once into LDS via CDNA5 async copies.

#define ROW   8192
#define NTHR  256
#define PER_T (ROW / NTHR)   // 32 floats / thread
#define NVEC  (PER_T / 4)    // 8 float4 / thread

// bias = bf16(2^4 - log2(448) + log2(2 - 2^(1-5)) - 1) = bf16(7.1468414) = 7.15625
#define FLEX_BIAS 7.15625f
#define MBITS     5.0f

typedef float v4f __attribute__((ext_vector_type(4)));

__global__ __launch_bounds__(NTHR) void fpq_rowquant_kernel(
    const float* __restrict__ x, float* __restrict__ out) {
  __shared__ float srow[ROW];          // 32 KB row stage
  __shared__ float swmax[NTHR / 32];   // per-wave partial maxima

  const int tid = threadIdx.x;
  const size_t rowbase = (size_t)blockIdx.x * (size_t)ROW;
  const float* gsrc = x + rowbase;
  float* gdst = out + rowbase;

  // ---- Stage row into LDS with async global->LDS copies (ASYNCcnt path) ----
#pragma unroll
  for (int k = 0; k < NVEC; ++k) {
    const int idx = (tid + k * NTHR) * 4;  // float index; 4 KB contiguous per k
    uint32_t lds_off = (uint32_t)(uintptr_t)(&srow[idx]);
    uint64_t gaddr = (uint64_t)(uintptr_t)(gsrc + idx);
    asm volatile("global_load_async_to_lds_b128 %0, %1, off"
                 :: "v"(lds_off), "v"(gaddr)
                 : "memory");
  }
  asm volatile("s_wait_asynccnt 0" ::: "memory");
  __syncthreads();

  // ---- Pass 1: row absmax from LDS ----
  float m = 0.0f;
#pragma unroll
  for (int k = 0; k < NVEC; ++k) {
    const int idx = (tid + k * NTHR) * 4;
    v4f v = *reinterpret_cast<const v4f*>(&srow[idx]);
    m = fmaxf(m, fmaxf(fmaxf(__builtin_fabsf(v.x), __builtin_fabsf(v.y)),
                       fmaxf(__builtin_fabsf(v.z), __builtin_fabsf(v.w))));
  }
  // wave32 butterfly reduction
#pragma unroll
  for (int off = 16; off > 0; off >>= 1)
    m = fmaxf(m, __shfl_xor(m, off, 32));
  if ((tid & 31) == 0) swmax[tid >> 5] = m;
  __syncthreads();
  float rmax = swmax[0];
#pragma unroll
  for (int w = 1; w < NTHR / 32; ++w) rmax = fmaxf(rmax, swmax[w]);

  // ---- Pass 2: quantize from LDS, nontemporal store to global ----
#pragma unroll
  for (int k = 0; k < NVEC; ++k) {
    const int idx = (tid + k * NTHR) * 4;
    v4f v = *reinterpret_cast<const v4f*>(&srow[idx]);
    v4f r;
#pragma unroll
    for (int j = 0; j < 4; ++j) {
      float xc = fminf(fmaxf(v[j], -rmax), rmax);
      // v_log_f32(0) -> -inf -> floor -> -inf -> fmax(.,1) = 1  (matches reference)
      float ls = floorf(__builtin_amdgcn_logf(__builtin_fabsf(xc)) + FLEX_BIAS);
      ls = fmaxf(ls, 1.0f);
      const float e = ls - MBITS - FLEX_BIAS;
      const float scale = __builtin_amdgcn_exp2f(e);
      const float inv_scale = __builtin_amdgcn_exp2f(-e);
      // default mode is RNE, so rintf == round-half-to-even (jnp.round)
      r[j] = rintf(xc * inv_scale) * scale;
    }
    __builtin_nontemporal_store(r, reinterpret_cast<v4f*>(&gdst[idx]));
  }
}

extern "C" void kernel_launch(void* const* d_in, const int* in_sizes, int n_in,
                              void* d_out, int out_size, void* d_ws, size_t ws_size,
                              hipStream_t stream) {
  (void)n_in; (void)d_ws; (void)ws_size; (void)out_size;
  const float* x = (const float*)d_in[0];
  float* out = (float*)d_out;
  const int rows = in_sizes[0] / ROW;  // 8192
  fpq_rowquant_kernel<<<rows, NTHR, 0, stream>>>(x, out);
}